// Net_skip_11390253269709
// MI455X (gfx1250) — compile-verified
//
#include <hip/hip_runtime.h>

typedef __attribute__((ext_vector_type(2))) float v2f;
typedef __attribute__((ext_vector_type(8))) float v8f;

// ---------------------------------------------------------------- utilities

__global__ void k_fill_zero(float* __restrict__ p, long long n) {
  long long i = (long long)blockIdx.x * blockDim.x + threadIdx.x;
  if (i < n) p[i] = 0.0f;
}

// inp4[i] = {x[i,0], x[i,1], y1[i], 0}; rows N..Npad-1 zeroed
__global__ void k_prep_inp4(const float* __restrict__ x, const float* __restrict__ y1,
                            float* __restrict__ inp4, int N, int Npad) {
  int i = blockIdx.x * blockDim.x + threadIdx.x;
  if (i < Npad) {
    float a = 0.f, b = 0.f, c = 0.f;
    if (i < N) { a = x[i * 2 + 0]; b = x[i * 2 + 1]; c = y1[i]; }
    inp4[i * 4 + 0] = a;
    inp4[i * 4 + 1] = b;
    inp4[i * 4 + 2] = c;
    inp4[i * 4 + 3] = 0.0f;
  }
}

// Pre-swizzle W[Kdim,ncols] (row-major) into WMMA B-fragment order:
// WB[((colTile*ksteps + step)*32 + lane)*2 + {0,1}] = W[ka][col], W[ka+1][col]
// with ka = step*4 + 2*(lane>>4), col = colTile*16 + (lane&15); zero-padded.
__global__ void k_prep_wb(const float* __restrict__ W, int Kdim, int ncols,
                          int ksteps, int nColTiles, float* __restrict__ WB) {
  int t = blockIdx.x * blockDim.x + threadIdx.x;
  int total = nColTiles * ksteps * 32;
  if (t >= total) return;
  int lane = t & 31;
  int step = (t >> 5) % ksteps;
  int colTile = (t >> 5) / ksteps;
  int half = lane >> 4, lid = lane & 15;
  int ka = step * 4 + half * 2;
  int col = colTile * 16 + lid;
  float v0 = (ka     < Kdim && col < ncols) ? W[ka * ncols + col]       : 0.0f;
  float v1 = (ka + 1 < Kdim && col < ncols) ? W[(ka + 1) * ncols + col] : 0.0f;
  WB[(size_t)t * 2 + 0] = v0;
  WB[(size_t)t * 2 + 1] = v1;
}

// deg[dst] += 1 (f32 atomic exact for counts < 2^24)
__global__ void k_deg(const long long* __restrict__ dst, float* __restrict__ deg,
                      long long E) {
  long long e = (long long)blockIdx.x * blockDim.x + threadIdx.x;
  if (e < E) {
    int d = (int)__builtin_nontemporal_load(&dst[e]);  // NT: don't pollute L2
    atomicAdd(&deg[d], 1.0f);
  }
}

// in-place deg -> rsqrt(deg + 1)   (self-loop)
__global__ void k_dinv(float* __restrict__ dinv, int N) {
  int i = blockIdx.x * blockDim.x + threadIdx.x;
  if (i < N) dinv[i] = rsqrtf(dinv[i] + 1.0f);
}

// ------------------------------------------------------- dense X@W via WMMA
// One wave computes a 16x16 tile of XW with KSTEPS back-to-back
// V_WMMA_F32_16X16X4_F32 ops. All loads unconditional (inputs pre-padded):
// per step: one b64 A-fragment load + one b64 B-fragment load + wmma.
template <int KSTEPS>
__global__ void k_matmul_wmma(const float* __restrict__ F, int ldF,
                              const float* __restrict__ WB,
                              float* __restrict__ XW, int nNodeTiles,
                              int nColTiles) {
  int gw = blockIdx.x * (blockDim.x >> 5) + (threadIdx.x >> 5);
  int nodeTile = gw / nColTiles;
  int colTile  = gw % nColTiles;
  if (nodeTile >= nNodeTiles) return;  // wave-uniform exit (EXEC stays full)

  int lane = threadIdx.x & 31;
  int half = lane >> 4;                // selects K pair within a 4-wide step
  int lid  = lane & 15;

  const float* Arow = F + (size_t)(nodeTile * 16 + lid) * ldF + half * 2;
  const float* Bp   = WB + ((size_t)colTile * KSTEPS * 32 + lane) * 2;

  v8f acc = {};
#pragma unroll
  for (int s = 0; s < KSTEPS; ++s) {
    v2f a = *(const v2f*)(Arow + s * 4);   // A[M=lid][ka], A[M=lid][ka+1]
    v2f b = *(const v2f*)(Bp + s * 64);    // W[ka][col],  W[ka+1][col]
    acc = __builtin_amdgcn_wmma_f32_16x16x4_f32(
        /*neg_a=*/false, a, /*neg_b=*/false, b,
        /*c_mod=*/(short)0, acc, /*reuse_a=*/false, /*reuse_b=*/false);
  }

  // D layout: lanes 0-15 -> M = r, lanes 16-31 -> M = 8 + r, N = lid
  float* outp = XW + (size_t)(nodeTile * 16 + half * 8) * 32 + colTile * 16 + lid;
#pragma unroll
  for (int r = 0; r < 8; ++r) outp[(size_t)r * 32] = acc[r];
}

// --------------------------------------------------- edge gather/scatter-add
// One wave per edge; lanes 0..28 move one 29-float row (single 128B line).
__global__ void k_edge_scatter29(const long long* __restrict__ src,
                                 const long long* __restrict__ dst,
                                 const float* __restrict__ dinv,
                                 const float* __restrict__ XW,
                                 float* __restrict__ AGG, long long E) {
  long long e = (long long)blockIdx.x * (blockDim.x >> 5) + (threadIdx.x >> 5);
  if (e >= E) return;
  int j = threadIdx.x & 31;
  int s = (int)__builtin_nontemporal_load(&src[e]);
  int d = (int)__builtin_nontemporal_load(&dst[e]);
  float c = dinv[s] * dinv[d];
  if (j < 29) {
    float v = XW[(long long)s * 32 + j] * c;
    atomicAdd(&AGG[(long long)d * 32 + j], v);
  }
}

// scalar (1-wide) variant for the last layer
__global__ void k_edge_scatter1(const long long* __restrict__ src,
                                const long long* __restrict__ dst,
                                const float* __restrict__ dinv,
                                const float* __restrict__ XW,
                                float* __restrict__ AGG1, long long E) {
  long long e = (long long)blockIdx.x * blockDim.x + threadIdx.x;
  if (e >= E) return;
  int s = (int)__builtin_nontemporal_load(&src[e]);
  int d = (int)__builtin_nontemporal_load(&dst[e]);
  float c = dinv[s] * dinv[d];
  atomicAdd(&AGG1[d], XW[(long long)s * 32] * c);
}

// ------------------------------------------------------------- finalization
// FEAT[i, j<29] = relu(AGG + XW*dinv^2 + b[j]); FEAT[i, 29..31] = inp4[i, 0..2]
__global__ void k_finalize_cat(const float* __restrict__ XW, const float* __restrict__ AGG,
                               const float* __restrict__ dinv, const float* __restrict__ bias,
                               const float* __restrict__ inp4,
                               float* __restrict__ FEAT, int N) {
  long long idx = (long long)blockIdx.x * blockDim.x + threadIdx.x;
  int i = (int)(idx >> 5);
  int j = (int)(idx & 31);
  if (i >= N) return;
  if (j < 29) {
    float di = dinv[i];
    float v = AGG[idx] + XW[idx] * (di * di) + bias[j];
    FEAT[idx] = fmaxf(v, 0.0f);
  } else {
    FEAT[idx] = inp4[i * 4 + (j - 29)];
  }
}

__global__ void k_finalize3(const float* __restrict__ XW, const float* __restrict__ AGG1,
                            const float* __restrict__ dinv, const float* __restrict__ b3,
                            float* __restrict__ out, int N) {
  int i = blockIdx.x * blockDim.x + threadIdx.x;
  if (i < N) {
    float di = dinv[i];
    out[i] = AGG1[i] + XW[(long long)i * 32] * (di * di) + b3[0];
  }
}

// -------------------------------------------------------------------- launch

static inline unsigned cdivu(long long a, long long b) {
  return (unsigned)((a + b - 1) / b);
}

extern "C" void kernel_launch(void* const* d_in, const int* in_sizes, int n_in,
                              void* d_out, int out_size, void* d_ws, size_t ws_size,
                              hipStream_t stream) {
  const float*     x   = (const float*)d_in[0];
  const float*     y1  = (const float*)d_in[1];
  const long long* ei  = (const long long*)d_in[2];
  const float*     W1  = (const float*)d_in[3];
  const float*     b1  = (const float*)d_in[4];
  const float*     W2  = (const float*)d_in[5];
  const float*     b2  = (const float*)d_in[6];
  const float*     W3  = (const float*)d_in[7];
  const float*     b3  = (const float*)d_in[8];
  float*           out = (float*)d_out;

  const int       N    = in_sizes[1];
  const int       Npad = (N + 15) & ~15;
  const long long E    = (long long)in_sizes[2] / 2;
  const long long* src = ei;       // edge_index[0]
  const long long* dst = ei + E;   // edge_index[1]

  // workspace carve-out (256B aligned)
  char* ws = (char*)d_ws;
  size_t off = 0;
  auto alignup = [](size_t v) { return (v + 255) & ~(size_t)255; };
  float* dinv = (float*)(ws + off); off = alignup(off + sizeof(float) * (size_t)N);
  float* INP4 = (float*)(ws + off); off = alignup(off + sizeof(float) * (size_t)Npad * 4);
  float* XW   = (float*)(ws + off); off = alignup(off + sizeof(float) * (size_t)Npad * 32);
  float* AGG  = (float*)(ws + off); off = alignup(off + sizeof(float) * (size_t)Npad * 32);
  float* FEAT = (float*)(ws + off); off = alignup(off + sizeof(float) * (size_t)Npad * 32);
  float* WB1  = (float*)(ws + off); off = alignup(off + sizeof(float) * 2 * 1 * 64);
  float* WB2  = (float*)(ws + off); off = alignup(off + sizeof(float) * 2 * 8 * 64);
  float* WB3  = (float*)(ws + off); off = alignup(off + sizeof(float) * 1 * 8 * 64);
  (void)ws_size; (void)n_in; (void)out_size;

  const int B = 256;
  const int nNodeTiles = Npad / 16;

  // prep: padded input, swizzled weights, degree, normalization
  k_prep_inp4<<<cdivu(Npad, B), B, 0, stream>>>(x, y1, INP4, N, Npad);
  k_prep_wb<<<cdivu(2 * 1 * 32, 64), 64, 0, stream>>>(W1, 3, 29, 1, 2, WB1);
  k_prep_wb<<<cdivu(2 * 8 * 32, 64), 64, 0, stream>>>(W2, 32, 29, 8, 2, WB2);
  k_prep_wb<<<cdivu(1 * 8 * 32, 64), 64, 0, stream>>>(W3, 32, 1, 8, 1, WB3);
  if (Npad > N)  // zero FEAT pad rows once (they are read by layer-2/3 WMMA)
    k_fill_zero<<<cdivu((long long)(Npad - N) * 32, B), B, 0, stream>>>(
        FEAT + (size_t)N * 32, (long long)(Npad - N) * 32);
  k_fill_zero<<<cdivu(N, B), B, 0, stream>>>(dinv, N);
  k_deg<<<cdivu(E, B), B, 0, stream>>>(dst, dinv, E);
  k_dinv<<<cdivu(N, B), B, 0, stream>>>(dinv, N);

  // ---- layer 1: INP4[Npad,4] @ W1 -> XW (29 cols, stride 32)
  {
    long long waves = (long long)nNodeTiles * 2;
    k_matmul_wmma<1><<<cdivu(waves * 32, B), B, 0, stream>>>(INP4, 4, WB1, XW,
                                                             nNodeTiles, 2);
    k_fill_zero<<<cdivu((long long)N * 32, B), B, 0, stream>>>(AGG, (long long)N * 32);
    k_edge_scatter29<<<cdivu(E * 32, B), B, 0, stream>>>(src, dst, dinv, XW, AGG, E);
    k_finalize_cat<<<cdivu((long long)N * 32, B), B, 0, stream>>>(XW, AGG, dinv, b1,
                                                                  INP4, FEAT, N);
  }
  // ---- layer 2: FEAT[Npad,32] @ W2 -> XW
  {
    long long waves = (long long)nNodeTiles * 2;
    k_matmul_wmma<8><<<cdivu(waves * 32, B), B, 0, stream>>>(FEAT, 32, WB2, XW,
                                                             nNodeTiles, 2);
    k_fill_zero<<<cdivu((long long)N * 32, B), B, 0, stream>>>(AGG, (long long)N * 32);
    k_edge_scatter29<<<cdivu(E * 32, B), B, 0, stream>>>(src, dst, dinv, XW, AGG, E);
    k_finalize_cat<<<cdivu((long long)N * 32, B), B, 0, stream>>>(XW, AGG, dinv, b2,
                                                                  INP4, FEAT, N);
  }
  // ---- layer 3: FEAT[Npad,32] @ W3 (1 col) -> XW col 0; scalar aggregate
  {
    long long waves = (long long)nNodeTiles;
    k_matmul_wmma<8><<<cdivu(waves * 32, B), B, 0, stream>>>(FEAT, 32, WB3, XW,
                                                             nNodeTiles, 1);
    k_fill_zero<<<cdivu(N, B), B, 0, stream>>>(AGG, N);
    k_edge_scatter1<<<cdivu(E, B), B, 0, stream>>>(src, dst, dinv, XW, AGG, E);
    k_finalize3<<<cdivu(N, B), B, 0, stream>>>(XW, AGG, dinv, b3, out, N);
  }
}